// PyG_FeatureAttention_70282844832171
// MI455X (gfx1250) — compile-verified
//
#include <hip/hip_runtime.h>
#include <hip/hip_bf16.h>

// ---------------------------------------------------------------------------
// Problem constants (match reference)
// ---------------------------------------------------------------------------
#define B_    256
#define L_    256
#define C_    128
#define HID_  128
#define H_    4
#define D_    32
#define N_    (B_ * C_)       // 32768 nodes
#define NEG_SLOPE 0.2f
#define LN_EPS    1e-5f

typedef __attribute__((ext_vector_type(2))) float v2f;
typedef __attribute__((ext_vector_type(8))) float v8f;
typedef __attribute__((ext_vector_type(4))) unsigned int v4ui;
typedef __attribute__((ext_vector_type(4))) int v4i;
typedef __attribute__((ext_vector_type(8))) int v8i;

#ifndef __has_builtin
#define __has_builtin(x) 0
#endif
#if __has_builtin(__builtin_amdgcn_tensor_load_to_lds) && \
    __has_builtin(__builtin_amdgcn_s_wait_tensorcnt)
#define USE_TDM 1
#else
#define USE_TDM 0
#endif

// fp32 WMMA 16x16x4:  D = A(16x4) * B(4x16) + C(16x16)
__device__ inline v8f wmma_f32(v2f a, v2f b, v8f c) {
    return __builtin_amdgcn_wmma_f32_16x16x4_f32(
        /*neg_a=*/false, a, /*neg_b=*/false, b,
        /*c_mod=*/(short)0, c, /*reuse_a=*/false, /*reuse_b=*/false);
}

// ---------------------------------------------------------------------------
// TDM staging: DMA a KC x 128 fp32 weight chunk (row stride ldb elements)
// into LDS laid out as [KC][132] (hardware pad: +4 DWORDs every 128 DWORDs).
// Issued by one wave per block; all operands are wave-uniform (SGPRs).
// ---------------------------------------------------------------------------
#define KC 32

#if USE_TDM
__device__ inline void tdm_load_B(const float* gsrc, unsigned lds_off,
                                  int rows_remaining, int ldb_elems)
{
    unsigned long long ga = (unsigned long long)(uintptr_t)gsrc;
    v4ui g0;
    g0.x = 1u;                                   // count=1, user descriptor
    g0.y = lds_off;                              // lds_addr (bytes)
    g0.z = (unsigned)(ga & 0xFFFFFFFFu);         // global_addr[31:0]
    g0.w = (unsigned)((ga >> 32) & 0x01FFFFFFu)  // global_addr[56:32]
           | (2u << 30);                         // type = 2 ("image")
    v8i g1;
    // mask=0 | data_size=4B | pad_enable | pad_interval=128dw | pad_amount=4dw
    g1[0] = (2 << 16) | (1 << 20) | (6 << 22) | (3 << 25);
    g1[1] = (128 << 16);                         // tensor_dim0 = 128 (lo16)
    g1[2] = (rows_remaining & 0xFFFF) << 16;     // tensor_dim1 (lo16)
    g1[3] = (128 << 16);                         // tile_dim0 = 128
    g1[4] = KC;                                  // tile_dim1 = KC, tile_dim2 = 0
    g1[5] = ldb_elems;                           // tensor_dim0_stride (lo32)
    g1[6] = 0;                                   // stride hi / dim1_stride lo
    g1[7] = 0;
    v4i z4 = {0, 0, 0, 0};
#if __clang_major__ >= 23
    v8i z8 = {0, 0, 0, 0, 0, 0, 0, 0};
    __builtin_amdgcn_tensor_load_to_lds(g0, g1, z4, z4, z8, 0);
#else
    __builtin_amdgcn_tensor_load_to_lds(g0, g1, z4, z4, 0);
#endif
}
#endif

// stage one KC x 128 chunk of B into Bs; call from ALL threads of the block.
__device__ inline void stage_B(float (*Bs)[132], const float* __restrict__ chunk,
                               int rows_remaining, int ldb_elems, int tid)
{
#if USE_TDM
    if (tid < 32) {   // wave 0 issues the DMA (uniform operands)
        tdm_load_B(chunk, (unsigned)(uintptr_t)(void*)&Bs[0][0],
                   rows_remaining, ldb_elems);
        __builtin_amdgcn_s_wait_tensorcnt(0);
    }
#else
    #pragma unroll
    for (int i = 0; i < 4; ++i) {
        int lin = tid + i * 256;          // 0..1023 float4 slots
        int row = lin >> 5;
        int c4  = (lin & 31) << 2;
        float4 v = *(const float4*)(chunk + (size_t)row * ldb_elems + c4);
        Bs[row][c4 + 0] = v.x; Bs[row][c4 + 1] = v.y;
        Bs[row][c4 + 2] = v.z; Bs[row][c4 + 3] = v.w;
    }
#endif
}

// ---------------------------------------------------------------------------
// GEMM kernels.  Per-wave tile: 16 rows x 128 cols (8 WMMA accumulators).
// Block = 256 threads = 8 waves -> 128 rows per block.
//
// Fragment mapping (wave32):
//   lane = half*16 + ln;  A: a[v]   = A[m0+ln][k + 2*half + v]
//                         B: b[v]   = B[k + 2*half + v][n0+ln]
//                         C: acc[r] = C[m0 + r + 8*half][n0+ln]
// ---------------------------------------------------------------------------

// Standard row-major A (lda == K), N == 128, ldc == 128.
__global__ __launch_bounds__(256) void gemm128_kernel(
    const float* __restrict__ A, const float* __restrict__ Bm,
    const float* __restrict__ bias, float* __restrict__ Cm, int K)
{
    __shared__ float Bs[KC][132];
    const int tid  = threadIdx.x;
    const int wave = tid >> 5;
    const int lane = tid & 31;
    const int half = lane >> 4;
    const int ln   = lane & 15;
    const int m0   = blockIdx.x * 128 + wave * 16;

    v8f acc[8] = {};

    for (int kc = 0; kc < K; kc += KC) {
        stage_B(Bs, Bm + (size_t)kc * 128, K - kc, 128, tid);
        __syncthreads();

        const float* arow = A + (size_t)(m0 + ln) * K + kc;
        #pragma unroll
        for (int kk = 0; kk < KC; kk += 4) {
            int k2 = kk + 2 * half;
            v2f a; a.x = arow[k2]; a.y = arow[k2 + 1];
            #pragma unroll
            for (int nt = 0; nt < 8; ++nt) {
                v2f b; b.x = Bs[k2][nt * 16 + ln]; b.y = Bs[k2 + 1][nt * 16 + ln];
                acc[nt] = wmma_f32(a, b, acc[nt]);
            }
        }
        __syncthreads();
    }

    #pragma unroll
    for (int nt = 0; nt < 8; ++nt) {
        int col = nt * 16 + ln;
        float bv = bias[col];
        #pragma unroll
        for (int r = 0; r < 8; ++r) {
            int row = m0 + r + 8 * half;
            Cm[(size_t)row * 128 + col] = acc[nt][r] + bv;
        }
    }
}

// Embedding GEMM: A is the (B,L,C) input read as nodes x L with stride C.
//   node n = b*C + c ; A[n][l] = X[(b*L + l)*C + c]
__global__ __launch_bounds__(256) void gemm_embed_kernel(
    const float* __restrict__ X, const float* __restrict__ Wemb,
    const float* __restrict__ bias, float* __restrict__ Hout)
{
    __shared__ float Bs[KC][132];
    const int tid  = threadIdx.x;
    const int wave = tid >> 5;
    const int lane = tid & 31;
    const int half = lane >> 4;
    const int ln   = lane & 15;
    const int m0   = blockIdx.x * 128 + wave * 16;
    const int node = m0 + ln;
    const float* xbase = X + (size_t)(node >> 7) * L_ * C_ + (node & 127);

    v8f acc[8] = {};

    for (int kc = 0; kc < L_; kc += KC) {
        stage_B(Bs, Wemb + (size_t)kc * 128, L_ - kc, 128, tid);
        __syncthreads();

        #pragma unroll
        for (int kk = 0; kk < KC; kk += 4) {
            int k2 = kc + kk + 2 * half;
            v2f a; a.x = xbase[(size_t)k2 * C_]; a.y = xbase[(size_t)(k2 + 1) * C_];
            #pragma unroll
            for (int nt = 0; nt < 8; ++nt) {
                v2f b; b.x = Bs[kk + 2 * half][nt * 16 + ln];
                b.y = Bs[kk + 2 * half + 1][nt * 16 + ln];
                acc[nt] = wmma_f32(a, b, acc[nt]);
            }
        }
        __syncthreads();
    }

    #pragma unroll
    for (int nt = 0; nt < 8; ++nt) {
        int col = nt * 16 + ln;
        float bv = bias[col];
        #pragma unroll
        for (int r = 0; r < 8; ++r) {
            int row = m0 + r + 8 * half;
            Hout[(size_t)row * 128 + col] = acc[nt][r] + bv;
        }
    }
}

// Projection GEMM: out = h (Nx128) @ proj_W (128x256) + b, stored transposed
// into (B, L, C):  d_out[(b*L + l)*C + c]  with node = b*C + c, col = l.
// grid.y in {0,1} picks the 128-wide N slab.
__global__ __launch_bounds__(256) void gemm_proj_kernel(
    const float* __restrict__ Hin, const float* __restrict__ Wp,
    const float* __restrict__ bias, float* __restrict__ Out)
{
    __shared__ float Bs[KC][132];
    const int tid   = threadIdx.x;
    const int wave  = tid >> 5;
    const int lane  = tid & 31;
    const int half  = lane >> 4;
    const int ln    = lane & 15;
    const int m0    = blockIdx.x * 128 + wave * 16;
    const int nbase = blockIdx.y * 128;

    v8f acc[8] = {};

    for (int kc = 0; kc < HID_; kc += KC) {
        stage_B(Bs, Wp + (size_t)kc * 256 + nbase, HID_ - kc, 256, tid);
        __syncthreads();

        const float* arow = Hin + (size_t)(m0 + ln) * HID_ + kc;
        #pragma unroll
        for (int kk = 0; kk < KC; kk += 4) {
            int k2 = kk + 2 * half;
            v2f a; a.x = arow[k2]; a.y = arow[k2 + 1];
            #pragma unroll
            for (int nt = 0; nt < 8; ++nt) {
                v2f b; b.x = Bs[k2][nt * 16 + ln]; b.y = Bs[k2 + 1][nt * 16 + ln];
                acc[nt] = wmma_f32(a, b, acc[nt]);
            }
        }
        __syncthreads();
    }

    #pragma unroll
    for (int nt = 0; nt < 8; ++nt) {
        int col = nbase + nt * 16 + ln;   // l index, 0..255
        float bv = bias[col];
        #pragma unroll
        for (int r = 0; r < 8; ++r) {
            int row = m0 + r + 8 * half;  // node
            Out[((size_t)(row >> 7) * L_ + col) * C_ + (row & 127)] = acc[nt][r] + bv;
        }
    }
}

// ---------------------------------------------------------------------------
// Edge-phase kernels
// ---------------------------------------------------------------------------
__device__ inline void edge_decode(long long e, const int* __restrict__ ei, int E,
                                   int& src, int& dst)
{
    const long long BE = (long long)B_ * E;
    if (e < BE) {
        int ee = (int)e;
        int g  = ee / E;
        int k  = ee - g * E;
        src = ei[k]     + g * C_;
        dst = ei[E + k] + g * C_;
    } else {
        int n = (int)(e - BE);
        src = n; dst = n;
    }
}

__device__ inline unsigned fenc(float f) {
    unsigned u = __float_as_uint(f);
    return (u & 0x80000000u) ? ~u : (u | 0x80000000u);
}
__device__ inline float fdec(unsigned k) {
    unsigned u = (k & 0x80000000u) ? (k & 0x7FFFFFFFu) : ~k;
    return __uint_as_float(u);
}
__device__ inline float lrelu(float v) { return v > 0.f ? v : NEG_SLOPE * v; }

// Pass 1: logits per (edge, head) + segment max over dst (sortable-uint atomicMax)
__global__ __launch_bounds__(256) void edge_logits_kernel(
    const int* __restrict__ ei, int E, long long NE,
    const float* __restrict__ xl, const float* __restrict__ xr,
    const float* __restrict__ att,
    float* __restrict__ logits, unsigned* __restrict__ mkey)
{
    long long idx = (long long)blockIdx.x * 256 + threadIdx.x;   // edge*H + h
    if (idx >= NE * H_) return;
    long long e = idx >> 2;
    int h = (int)(idx & 3);
    int src, dst;
    edge_decode(e, ei, E, src, dst);

    const float4* pl = (const float4*)(xl + (size_t)src * HID_ + h * D_);
    const float4* pr = (const float4*)(xr + (size_t)dst * HID_ + h * D_);
    const float4* pa = (const float4*)(att + h * D_);
    float acc = 0.f;
    #pragma unroll
    for (int i = 0; i < 8; ++i) {
        float4 a4 = pa[i], l4 = pl[i], r4 = pr[i];
        acc += a4.x * lrelu(l4.x + r4.x);
        acc += a4.y * lrelu(l4.y + r4.y);
        acc += a4.z * lrelu(l4.z + r4.z);
        acc += a4.w * lrelu(l4.w + r4.w);
    }
    logits[idx] = acc;
    atomicMax(&mkey[(size_t)dst * H_ + h], fenc(acc));
}

// Pass 2: p = exp(logit - m[dst]) ; s[dst] += p
__global__ __launch_bounds__(256) void edge_softmax_kernel(
    const int* __restrict__ ei, int E, long long NE,
    const float* __restrict__ logits, const unsigned* __restrict__ mkey,
    float* __restrict__ p, float* __restrict__ s)
{
    long long idx = (long long)blockIdx.x * 256 + threadIdx.x;
    if (idx >= NE * H_) return;
    long long e = idx >> 2;
    int h = (int)(idx & 3);
    int src, dst;
    edge_decode(e, ei, E, src, dst);
    float m  = fdec(mkey[(size_t)dst * H_ + h]);
    float pe = __expf(logits[idx] - m);
    p[idx] = pe;
    atomicAdd(&s[(size_t)dst * H_ + h], pe);
}

// Pass 3: wave per edge; lane d scatters xl[src][h*32+d]*alpha[h] into newh[dst].
// Also accumulates the dense attention map for the last layer.
__global__ __launch_bounds__(256) void edge_aggregate_kernel(
    const int* __restrict__ ei, int E, long long NE,
    const float* __restrict__ xl, const float* __restrict__ p,
    const float* __restrict__ s,
    float* __restrict__ newh, float* __restrict__ attn, int do_attn)
{
    long long e = (long long)blockIdx.x * 8 + (threadIdx.x >> 5);
    if (e >= NE) return;
    int lane = threadIdx.x & 31;
    int src, dst;
    edge_decode(e, ei, E, src, dst);

    float alpha[H_];
    float amean = 0.f;
    #pragma unroll
    for (int h = 0; h < H_; ++h) {
        float pe = p[e * H_ + h];
        float se = s[(size_t)dst * H_ + h];
        alpha[h] = pe / (se + 1e-16f);
        amean += alpha[h];
    }
    amean *= (1.f / H_);

    #pragma unroll
    for (int h = 0; h < H_; ++h) {
        float v = xl[(size_t)src * HID_ + h * D_ + lane] * alpha[h];
        atomicAdd(&newh[(size_t)dst * HID_ + h * D_ + lane], v);
    }
    if (do_attn && lane == 0) {
        int b = src >> 7;   // src / C
        // attn_map[b, dst%C, src%C] += mean_h(alpha)   (to_dense_adj + transpose)
        atomicAdd(&attn[((size_t)b * C_ + (dst & 127)) * C_ + (src & 127)], amean);
    }
}

// Fused: new_h = elu(agg + gat_bias); h += new_h; h = layer_norm(h). Wave/node.
__global__ __launch_bounds__(256) void fuse_ln_kernel(
    float* __restrict__ h, const float* __restrict__ newh,
    const float* __restrict__ gbias, const float* __restrict__ gamma,
    const float* __restrict__ beta)
{
    int node = blockIdx.x * 8 + (threadIdx.x >> 5);
    int lane = threadIdx.x & 31;

    float v[4];
    float lsum = 0.f;
    #pragma unroll
    for (int i = 0; i < 4; ++i) {
        int c = i * 32 + lane;
        float nh = newh[(size_t)node * HID_ + c] + gbias[c];
        nh = nh > 0.f ? nh : (__expf(nh) - 1.f);       // elu
        float hv = h[(size_t)node * HID_ + c] + nh;    // residual
        v[i] = hv;
        lsum += hv;
    }
    #pragma unroll
    for (int off = 16; off > 0; off >>= 1) lsum += __shfl_xor(lsum, off, 32);
    float mu = lsum * (1.f / HID_);

    float vs = 0.f;
    #pragma unroll
    for (int i = 0; i < 4; ++i) { float d = v[i] - mu; vs += d * d; }
    #pragma unroll
    for (int off = 16; off > 0; off >>= 1) vs += __shfl_xor(vs, off, 32);
    float rinv = rsqrtf(vs * (1.f / HID_) + LN_EPS);

    #pragma unroll
    for (int i = 0; i < 4; ++i) {
        int c = i * 32 + lane;
        h[(size_t)node * HID_ + c] = (v[i] - mu) * rinv * gamma[c] + beta[c];
    }
}

// ---------------------------------------------------------------------------
// Host-side orchestration
// ---------------------------------------------------------------------------
extern "C" void kernel_launch(void* const* d_in, const int* in_sizes, int n_in,
                              void* d_out, int out_size, void* d_ws, size_t ws_size,
                              hipStream_t stream)
{
    const float* x        = (const float*)d_in[0];
    const int*   ei       = (const int*)d_in[1];
    const float* emb_W    = (const float*)d_in[2];
    const float* emb_b    = (const float*)d_in[3];
    const float* lin_l_W  = (const float*)d_in[4];
    const float* lin_l_b  = (const float*)d_in[5];
    const float* lin_r_W  = (const float*)d_in[6];
    const float* lin_r_b  = (const float*)d_in[7];
    const float* att      = (const float*)d_in[8];
    const float* gat_bias = (const float*)d_in[9];
    const float* ln_gamma = (const float*)d_in[10];
    const float* ln_beta  = (const float*)d_in[11];
    const float* proj_W   = (const float*)d_in[12];
    const float* proj_b   = (const float*)d_in[13];

    const int E = in_sizes[1] / 2;
    const long long NE = (long long)B_ * E + (long long)N_;

    // workspace carve-up (floats)
    float* h      = (float*)d_ws;
    float* xl     = h  + (size_t)N_ * HID_;
    float* xr     = xl + (size_t)N_ * HID_;
    float* nh     = xr + (size_t)N_ * HID_;
    float* logits = nh + (size_t)N_ * HID_;
    float* pbuf   = logits + (size_t)NE * H_;
    float* sbuf   = pbuf   + (size_t)NE * H_;
    unsigned* mkey = (unsigned*)(sbuf + (size_t)N_ * H_);

    float* out  = (float*)d_out;
    float* attn = out + (size_t)B_ * L_ * C_;

    hipMemsetAsync(attn, 0, sizeof(float) * (size_t)B_ * C_ * C_, stream);

    // 1) embedding GEMM
    gemm_embed_kernel<<<N_ / 128, 256, 0, stream>>>(x, emb_W, emb_b, h);

    const unsigned ehBlocks = (unsigned)((NE * H_ + 255) / 256);
    const unsigned eBlocks  = (unsigned)((NE + 7) / 8);

    for (int layer = 0; layer < 2; ++layer) {
        hipMemsetAsync(nh,   0, sizeof(float)    * (size_t)N_ * HID_, stream);
        hipMemsetAsync(sbuf, 0, sizeof(float)    * (size_t)N_ * H_,   stream);
        hipMemsetAsync(mkey, 0, sizeof(unsigned) * (size_t)N_ * H_,   stream);

        gemm128_kernel<<<N_ / 128, 256, 0, stream>>>(
            h, lin_l_W + (size_t)layer * HID_ * HID_, lin_l_b + layer * HID_, xl, HID_);
        gemm128_kernel<<<N_ / 128, 256, 0, stream>>>(
            h, lin_r_W + (size_t)layer * HID_ * HID_, lin_r_b + layer * HID_, xr, HID_);

        edge_logits_kernel<<<ehBlocks, 256, 0, stream>>>(
            ei, E, NE, xl, xr, att + layer * H_ * D_, logits, mkey);
        edge_softmax_kernel<<<ehBlocks, 256, 0, stream>>>(
            ei, E, NE, logits, mkey, pbuf, sbuf);
        edge_aggregate_kernel<<<eBlocks, 256, 0, stream>>>(
            ei, E, NE, xl, pbuf, sbuf, nh, attn, layer == 1);

        fuse_ln_kernel<<<N_ / 8, 256, 0, stream>>>(
            h, nh, gat_bias + layer * HID_, ln_gamma + layer * HID_,
            ln_beta + layer * HID_);
    }

    // 4) projection GEMM with transposed (B,L,C) store
    gemm_proj_kernel<<<dim3(N_ / 128, 2), 256, 0, stream>>>(h, proj_W, proj_b, out);
}